// S4DKernel_50148038148621
// MI455X (gfx1250) — compile-verified
//
#include <hip/hip_runtime.h>
#include <math.h>

typedef float v2f __attribute__((ext_vector_type(2)));
typedef float v8f __attribute__((ext_vector_type(8)));

#define NH      32      // half state dim (complex poles per head)
#define T_TILE  64      // output columns per tile-row  (N dim of GEMM)
#define B_TILE  128     // output tile rows = L / T_TILE (M dim of GEMM)
#define KDIM    64      // 2 * NH (re/im interleaved)    (K dim of GEMM)
#define LDA     68      // padded LDS row stride (floats) -> conflict-free frag loads
#define LDB     68

// K[h, b*64 + t] = sum_k A2[b,k] * B2[k,t]
//   A2[b,2n]   =  2*Re(C'[n] * W[n]^(64b))      B2[2n,t]   = Re(W[n]^t)
//   A2[b,2n+1] = -2*Im(C'[n] * W[n]^(64b))      B2[2n+1,t] = Im(W[n]^t)
// with W = exp(dtA), C' = C * (exp(dtA)-1)/A.
__global__ __launch_bounds__(128)
void s4d_wmma_kernel(const float* __restrict__ C_re,
                     const float* __restrict__ C_im,
                     const float* __restrict__ log_dt,
                     const float* __restrict__ log_A_real,
                     const float* __restrict__ A_imag,
                     float* __restrict__ out, int L)
{
    __shared__ float sA[B_TILE][LDA];   // A2 : 128 x 64 (+pad)
    __shared__ float sB[KDIM][LDB];     // B2 :  64 x 64 (+pad)
    __shared__ float s_dre[NH], s_dim[NH], s_c2re[NH], s_c2im[NH];

    const int h   = blockIdx.x;
    const int tid = threadIdx.x;

    // ---------------- Phase 1: per-pole discretized coefficients ----------
    if (tid < NH) {
        const int n = tid;
        const float dt  = expf(log_dt[h]);
        const float aim = A_imag[(size_t)h * NH + n];
        const float are = -expf(log_A_real[(size_t)h * NH + n]);
        const float dre = are * dt;          // Re(dtA)
        const float dim = aim * dt;          // Im(dtA)
        // exp(dtA) - 1
        float sn, cs;
        const float er = expf(dre);
        sincosf(dim, &sn, &cs);
        const float em1_re = er * cs - 1.0f;
        const float em1_im = er * sn;
        // (exp(dtA)-1) / A   (A = are + i*aim)
        const float inv_den = 1.0f / (are * are + aim * aim);
        const float t_re = (em1_re * are + em1_im * aim) * inv_den;
        const float t_im = (em1_im * are - em1_re * aim) * inv_den;
        // 2 * C * t   (fold the final factor of 2 in here)
        const float cr = C_re[(size_t)h * NH + n];
        const float ci = C_im[(size_t)h * NH + n];
        s_dre[n]  = dre;
        s_dim[n]  = dim;
        s_c2re[n] = 2.0f * (cr * t_re - ci * t_im);
        s_c2im[n] = 2.0f * (cr * t_im + ci * t_re);
    }
    __syncthreads();

    // ---------------- Phase 2: build factor matrices in LDS ---------------
    // B2[k,t] = W^t   (32*64 = 2048 complex entries, 16 per thread)
    for (int i = tid; i < NH * T_TILE; i += 128) {
        const int n = i >> 6;
        const int t = i & 63;
        const float x = (float)t;
        const float mag = expf(s_dre[n] * x);
        float sn, cs;
        sincosf(s_dim[n] * x, &sn, &cs);
        sB[2 * n    ][t] = mag * cs;
        sB[2 * n + 1][t] = mag * sn;
    }
    // A2[b,k] = 2*C' * W^(64b)   (128*32 = 4096 complex entries, 32 per thread)
    for (int i = tid; i < B_TILE * NH; i += 128) {
        const int b = i >> 5;
        const int n = i & 31;
        const float x = (float)(b * T_TILE);
        const float mag = expf(s_dre[n] * x);
        float sn, cs;
        sincosf(s_dim[n] * x, &sn, &cs);
        const float wr = mag * cs, wi = mag * sn;
        const float gr = s_c2re[n] * wr - s_c2im[n] * wi;
        const float gi = s_c2re[n] * wi + s_c2im[n] * wr;
        sA[b][2 * n    ] = gr;
        sA[b][2 * n + 1] = -gi;
    }
    __syncthreads();

    // ---------------- Phase 3: WMMA GEMM 128x64x64 per head ---------------
    const int wave    = tid >> 5;        // 4 waves: each owns one 16-col strip
    const int lane    = tid & 31;
    const int r       = lane & 15;       // row (A/M) or col (B,D/N) within tile
    const int halfsel = lane >> 4;       // 0: lanes 0-15, 1: lanes 16-31
    const int ncol    = wave * 16 + r;   // output column within T_TILE

    // Preload this wave's 16 B fragments (reused across all 8 M tiles).
    // B 4x16 f32 frag: VGPR0 = row K0 (lanes 0-15) / K0+2 (lanes 16-31),
    //                  VGPR1 = row K0+1 / K0+3; column = lane&15.
    v2f bfrag[16];
#pragma unroll
    for (int kk = 0; kk < 16; ++kk) {
        const int krow = 4 * kk + 2 * halfsel;
        bfrag[kk][0] = sB[krow    ][ncol];
        bfrag[kk][1] = sB[krow + 1][ncol];
    }

    float* const obase = out + (size_t)h * L;
    for (int m = 0; m < 8; ++m) {
        v8f acc = {};
        const float* arow = &sA[m * 16 + r][0];
#pragma unroll
        for (int kk = 0; kk < 16; ++kk) {
            // A 16x4 f32 frag: VGPR0 = K0 (lanes 0-15) / K0+2 (lanes 16-31),
            //                  VGPR1 = K0+1 / K0+3; row M = lane&15.
            const int kc = 4 * kk + 2 * halfsel;
            v2f afrag;
            afrag[0] = arow[kc];
            afrag[1] = arow[kc + 1];
            acc = __builtin_amdgcn_wmma_f32_16x16x4_f32(
                /*neg_a=*/false, afrag, /*neg_b=*/false, bfrag[kk],
                /*c_mod=*/(short)0, acc, /*reuse_a=*/false, /*reuse_b=*/false);
        }
        // D 16x16 f32: VGPR j -> M = j + 8*halfsel, N = lane&15
#pragma unroll
        for (int j = 0; j < 8; ++j) {
            const int mrow = m * 16 + j + 8 * halfsel;
            obase[mrow * T_TILE + ncol] = acc[j];
        }
    }
}

extern "C" void kernel_launch(void* const* d_in, const int* in_sizes, int n_in,
                              void* d_out, int out_size, void* d_ws, size_t ws_size,
                              hipStream_t stream) {
    const float* C_re       = (const float*)d_in[0];
    const float* C_im       = (const float*)d_in[1];
    const float* log_dt     = (const float*)d_in[2];
    const float* log_A_real = (const float*)d_in[3];
    const float* A_imag     = (const float*)d_in[4];
    const int H = in_sizes[2];          // 512 (log_dt is (H,))
    const int L = out_size / H;         // 8192
    s4d_wmma_kernel<<<H, 128, 0, stream>>>(C_re, C_im, log_dt, log_A_real,
                                           A_imag, (float*)d_out, L);
}